// AttentionOverlapLoss_52673478918728
// MI455X (gfx1250) — compile-verified
//
#include <hip/hip_runtime.h>
#include <hip/hip_bf16.h>

// ---------------------------------------------------------------------------
// AttentionOverlapLoss for MI455X (gfx1250, wave32)
//
// Bandwidth-bound single-pass reduction:
//   per image: mn, mx, sumT = sum(x), sumIn = sum(x inside clamped bbox)
//   loss_i computed in closed form (identical math to reference), then mean.
//
// CDNA5-specific paths used:
//   - V_WMMA_F32_16X16X4_F32 as a cross-lane sum-reduction engine
//     (A = per-lane partials, B = all-ones -> D rows are pair sums)
//   - global_prefetch_b8 via __builtin_prefetch for the HBM stream
//
// Note: bboxes are read as int32 per the harness contract
// ("integer -> const int*"; JAX default config also downcasts int64->int32).
// ---------------------------------------------------------------------------

typedef __attribute__((ext_vector_type(2))) float v2f;
typedef __attribute__((ext_vector_type(8))) float v8f;

#define HW     224
#define NPIX   (HW * HW)     // 50176 elements / image
#define NQ     (NPIX / 4)    // 12544 float4 / image
#define QROW   (HW / 4)      // 56 float4 per row
#define BLK    256           // 8 wave32s per block
#define ITERS  (NQ / BLK)    // 49 float4 per thread (exact)

// Full-precision wave32 sum using the f32 WMMA (16x16x4).
// A layout (ISA 7.12.2): lane L<16 -> (M=L, K=0..1 in v[0..1]);
//                        lane L>=16 -> (M=L-16, K=2..3 in v[0..1]).
// With B = all-ones, D(m, n) = v_m + v_{m+16} for every column n.
// D layout: lane L<16 VGPR r = D(r, L); lane L>=16 VGPR r = D(8+r, L-16).
// So (sum of 8 D regs) + shfl_xor(·,16) = full wave sum, broadcast to all lanes.
__device__ __forceinline__ float wave_sum_wmma(float v) {
    v2f a; a.x = v;    a.y = 0.0f;
    v2f b; b.x = 1.0f; b.y = 1.0f;
    v8f c = {};
    c = __builtin_amdgcn_wmma_f32_16x16x4_f32(
            /*neg_a=*/false, a, /*neg_b=*/false, b,
            /*c_mod=*/(short)0, c, /*reuse_a=*/false, /*reuse_b=*/false);
    float t = c[0] + c[1] + c[2] + c[3] + c[4] + c[5] + c[6] + c[7];
    t += __shfl_xor(t, 16);
    return t;
}

__device__ __forceinline__ float wave_min(float v) {
    #pragma unroll
    for (int m = 16; m >= 1; m >>= 1) v = fminf(v, __shfl_xor(v, m));
    return v;
}
__device__ __forceinline__ float wave_max(float v) {
    #pragma unroll
    for (int m = 16; m >= 1; m >>= 1) v = fmaxf(v, __shfl_xor(v, m));
    return v;
}

__global__ void __launch_bounds__(BLK)
attn_loss_stage1(const float* __restrict__ attn,
                 const int* __restrict__ bboxes,
                 float* __restrict__ per_image_loss) {
    const int img = blockIdx.x;
    const int tid = threadIdx.x;

    const float4* __restrict__ A =
        reinterpret_cast<const float4*>(attn) + (size_t)img * NQ;

    // Clamped bbox (int32 per harness contract).
    const int* bb = bboxes + (size_t)img * 4;
    const int x1 = min(max(bb[0], 0), HW - 1);
    const int y1 = min(max(bb[1], 0), HW - 1);
    const int x2 = min(max(bb[2], 0), HW - 1);
    const int y2 = min(max(bb[3], 0), HW - 1);

    float sT = 0.0f, sI = 0.0f;
    float mn = 3.402823466e38f, mx = -3.402823466e38f;

    #pragma unroll 7
    for (int i = 0; i < ITERS; ++i) {
        const int idx = tid + i * BLK;
        // Prefetch the HBM stream ~32KB ahead (emits global_prefetch_b8).
        __builtin_prefetch(reinterpret_cast<const char*>(&A[idx]) + 32768, 0, 0);
        const float4 v = A[idx];

        sT += (v.x + v.y) + (v.z + v.w);
        mn = fminf(mn, fminf(fminf(v.x, v.y), fminf(v.z, v.w)));
        mx = fmaxf(mx, fmaxf(fmaxf(v.x, v.y), fmaxf(v.z, v.w)));

        const int r  = idx / QROW;           // row (0..223)
        const int c0 = (idx - r * QROW) * 4; // first column of this float4
        if (r >= y1 && r <= y2) {
            float p = 0.0f;
            p += (c0     >= x1 && c0     <= x2) ? v.x : 0.0f;
            p += (c0 + 1 >= x1 && c0 + 1 <= x2) ? v.y : 0.0f;
            p += (c0 + 2 >= x1 && c0 + 2 <= x2) ? v.z : 0.0f;
            p += (c0 + 3 >= x1 && c0 + 3 <= x2) ? v.w : 0.0f;
            sI += p;
        }
    }

    // Wave-level reductions (EXEC all-ones here: no divergence at block level).
    sT = wave_sum_wmma(sT);
    sI = wave_sum_wmma(sI);
    mn = wave_min(mn);
    mx = wave_max(mx);

    __shared__ float smT[8], smI[8], smMn[8], smMx[8];
    const int wave = tid >> 5;
    const int lane = tid & 31;
    if (lane == 0) { smT[wave] = sT; smI[wave] = sI; smMn[wave] = mn; smMx[wave] = mx; }
    __syncthreads();

    if (tid == 0) {
        float tT = 0.0f, tI = 0.0f;
        float tMn = 3.402823466e38f, tMx = -3.402823466e38f;
        #pragma unroll
        for (int w = 0; w < BLK / 32; ++w) {
            tT += smT[w]; tI += smI[w];
            tMn = fminf(tMn, smMn[w]); tMx = fmaxf(tMx, smMx[w]);
        }
        const float eps   = 1e-8f;
        const float denom = tMx - tMn + eps;
        const float Nf    = (float)NPIX;
        const int   bw    = x2 - x1 + 1;
        const int   bh    = y2 - y1 + 1;
        const float Nin   = (bw > 0 && bh > 0) ? (float)(bw * bh) : 0.0f;

        const float S      = (tT - Nf * tMn) / denom;                // sum of min-max-normalized map
        const float outSum = ((tT - tI) - (Nf - Nin) * tMn) / denom; // its sum outside the box
        per_image_loss[img] = outSum / (S + eps);
    }
}

__global__ void __launch_bounds__(BLK)
attn_loss_stage2(const float* __restrict__ per_image_loss,
                 float* __restrict__ out, int nimg) {
    float s = 0.0f;
    for (int i = threadIdx.x; i < nimg; i += BLK) s += per_image_loss[i];

    s = wave_sum_wmma(s);

    __shared__ float sm[8];
    const int wave = threadIdx.x >> 5;
    const int lane = threadIdx.x & 31;
    if (lane == 0) sm[wave] = s;
    __syncthreads();

    if (threadIdx.x == 0) {
        float t = 0.0f;
        #pragma unroll
        for (int w = 0; w < BLK / 32; ++w) t += sm[w];
        out[0] = t / (float)nimg;
    }
}

extern "C" void kernel_launch(void* const* d_in, const int* in_sizes, int n_in,
                              void* d_out, int out_size, void* d_ws, size_t ws_size,
                              hipStream_t stream) {
    const float* attn   = (const float*)d_in[0];
    const int*   bboxes = (const int*)d_in[1];
    float*       out    = (float*)d_out;
    float*       ws     = (float*)d_ws;   // per-image losses

    const int nimg = in_sizes[1] / 4;     // bboxes are (B, 4)

    attn_loss_stage1<<<nimg, BLK, 0, stream>>>(attn, bboxes, ws);
    attn_loss_stage2<<<1, BLK, 0, stream>>>(ws, out, nimg);
}